// LSTMCell_32590211842356
// MI455X (gfx1250) — compile-verified
//
#include <hip/hip_runtime.h>

// Packed-sequence LSTM on gfx1250 (MI455X), fp32 WMMA (V_WMMA_F32_16X16X4_F32).
//
// Path A (workspace fits total*1024 floats):
//   1) pregemm: G[total,1024] = data @ w_ih^T + (b_ih+b_hh)  -- fully parallel
//      across the whole GPU (~98k 16x16 tiles).
//   2) lstm_rec: 16 WGs x 512 threads; WG r owns batch rows [16r,16r+16).
//      Per step only the recurrent h @ w_hh^T GEMM runs (256 wmma / wave),
//      with the step's G tile loaded early and folded in at the elementwise
//      stage so its latency hides under the GEMM.
// Path B (fallback): fused kernel computing both GEMMs per step.
//
// The LSTM recurrence is row-local (gates[b,:] needs only h[b,:]), so batch
// rows partition across WGs with zero inter-WG sync; packed early-exit maps
// to a uniform per-WG break since lengths are sorted.
//
// Elementwise stage targets the gfx1250 hardware tanh (TRANS32) when the
// builtin exists; otherwise falls back to hardware exp (v_exp_f32).

typedef __attribute__((ext_vector_type(2))) float v2f;
typedef __attribute__((ext_vector_type(8))) float v8f;

#define TSTEPS 1024
#define HID    256
#define INDIM  256
#define G_STRIDE 1024  // 4*HID

static __device__ __forceinline__ v8f wmma4(v2f a, v2f b, v8f c) {
  // D(16x16) = A(16x4) * B(4x16) + C, fp32
  return __builtin_amdgcn_wmma_f32_16x16x4_f32(
      false, a, false, b, (short)0, c, false, false);
}

static __device__ __forceinline__ float fast_tanhf(float x) {
#if __has_builtin(__builtin_amdgcn_tanhf)
  return __builtin_amdgcn_tanhf(x);           // V_TANH_F32 (TRANS32)
#elif __has_builtin(__builtin_amdgcn_tanh_f32)
  return __builtin_amdgcn_tanh_f32(x);        // V_TANH_F32 (TRANS32)
#else
  // tanh(x) = 1 - 2/(exp(2x)+1); __expf lowers to v_exp_f32
  return 1.0f - 2.0f / (__expf(2.0f * x) + 1.0f);
#endif
}

static __device__ __forceinline__ float sigf(float x) {
  // sigmoid via tanh: one transcendental per gate
  return 0.5f + 0.5f * fast_tanhf(0.5f * x);
}

// ---------------------------------------------------------------------------
// Kernel 1 (path A): G = data @ w_ih^T + bias, tiled 16x16, K=256.
// Grid: ((total+15)/16, 8) x 256 threads; wave w of block (bx,by) computes
// output tile rows [16*bx,16*bx+16) x cols [16*(8*by+w), +16).
// ---------------------------------------------------------------------------
__global__ __launch_bounds__(256) void lstm_pregemm(
    const float* __restrict__ data, const float* __restrict__ w_ih,
    const float* __restrict__ b_ih, const float* __restrict__ b_hh,
    float* __restrict__ G, long long total)
{
  __shared__ float a_s[16 * INDIM];

  const long long row0 = (long long)blockIdx.x * 16;
  const int tid  = threadIdx.x;
  const int wave = tid >> 5;
  const int lane = tid & 31;
  const int ln16 = lane & 15;
  const int half = lane >> 4;

  // cooperative A-tile load (clamp tail rows; extra rows are never stored)
  {
    const int m  = tid >> 4;
    const int c0 = (tid & 15) * 16;
    long long row = row0 + m;
    if (row >= total) row = total - 1;
    const float4* src = (const float4*)(data + row * INDIM + c0);
    float4* dst = (float4*)&a_s[m * INDIM + c0];
    dst[0] = src[0]; dst[1] = src[1]; dst[2] = src[2]; dst[3] = src[3];
  }
  __syncthreads();

  const int n0 = (blockIdx.y * 8 + wave) * 16;  // output column base, 0..1008
  const float bv = b_ih[n0 + ln16] + b_hh[n0 + ln16];
  v8f acc;
  #pragma unroll
  for (int v = 0; v < 8; ++v) acc[v] = bv;

  const float* wb = w_ih + (size_t)(n0 + ln16) * INDIM + 2 * half;
  const float* ab = &a_s[ln16 * INDIM + 2 * half];
  #pragma unroll 4
  for (int k = 0; k < INDIM; k += 4) {
    v2f a = *(const v2f*)(ab + k);
    v2f b = *(const v2f*)(wb + k);
    acc = wmma4(a, b, acc);
  }

  const int col = n0 + ln16;
  float* gbase = G + row0 * G_STRIDE + col;
  if (row0 + 16 <= total) {  // fast path: whole row tile in bounds
    #pragma unroll
    for (int v = 0; v < 8; ++v)
      __builtin_nontemporal_store(acc[v],
                                  gbase + (v + 8 * half) * (long long)G_STRIDE);
  } else {
    #pragma unroll
    for (int v = 0; v < 8; ++v) {
      const long long m = v + 8 * half;
      if (row0 + m < total)
        __builtin_nontemporal_store(acc[v], gbase + m * (long long)G_STRIDE);
    }
  }
}

// ---------------------------------------------------------------------------
// Kernel 2 (path A): recurrent part. 16 WGs x 512 threads (16 wave32).
// Wave j owns H columns [16j,16j+16) and its 4 gate tiles (i,f,g,o).
// ---------------------------------------------------------------------------
__global__ __launch_bounds__(512, 1) void lstm_rec(
    const float* __restrict__ G, const float* __restrict__ w_hh,
    const int* __restrict__ batch_sizes,
    float* __restrict__ packed_out, float* __restrict__ h_fin,
    float* __restrict__ c_fin, long long total)
{
  __shared__ float h_s[16 * HID];

  const int r    = blockIdx.x;       // batch rows 16r..16r+15
  const int tid  = threadIdx.x;
  const int j    = tid >> 5;         // wave id == H-column tile, 0..15
  const int lane = tid & 31;
  const int ln16 = lane & 15;
  const int half = lane >> 4;
  const int col  = j * 16 + ln16;

  for (int i = tid; i < 16 * HID; i += 512) h_s[i] = 0.0f;

  v8f cst;
  #pragma unroll
  for (int v = 0; v < 8; ++v) cst[v] = 0.0f;

  // B-fragment base pointers: w_hh row = g*256 + col, k offset 2*half
  const float* wh = w_hh + (size_t)col * HID + 2 * half;

  long long off = 0;
  __syncthreads();

  for (int t = 0; t < TSTEPS; ++t) {
    const int bs_t = batch_sizes[t];
    const int act  = bs_t - r * 16;
    if (act <= 0) break;                    // uniform per WG (sorted lengths)
    const int nact = act > 16 ? 16 : act;

    // Issue G-tile loads early (consumed only after the GEMM below).
    float gv[4][8];
    #pragma unroll
    for (int v = 0; v < 8; ++v) {
      const int m = v + 8 * half;
      const long long row = (m < nact) ? (off + r * 16 + m) : 0;
      const float* gp = G + row * G_STRIDE + col;
      #pragma unroll
      for (int g = 0; g < 4; ++g)
        gv[g][v] = __builtin_nontemporal_load(gp + g * HID);
    }

    // gates_h = h @ w_hh^T  (zero-init; G already holds x-part + bias)
    v8f acc[4];
    #pragma unroll
    for (int g = 0; g < 4; ++g)
      #pragma unroll
      for (int v = 0; v < 8; ++v) acc[g][v] = 0.0f;

    const float* hs = &h_s[ln16 * HID + 2 * half];
    #pragma unroll 2
    for (int k = 0; k < HID; k += 4) {
      v2f a = *(const v2f*)(hs + k);
      #pragma unroll
      for (int g = 0; g < 4; ++g) {
        v2f b = *(const v2f*)(wh + (size_t)g * HID * HID + k);
        acc[g] = wmma4(a, b, acc[g]);
      }
    }
    __syncthreads();  // all h_s reads complete before masked writes

    // elementwise LSTM cell update in C-fragment layout
    float* outrow = packed_out + (off + r * 16) * (long long)HID + col;
    if (nact == 16) {  // common case: scalar branch, unconditional updates
      #pragma unroll
      for (int v = 0; v < 8; ++v) {
        const int m = v + 8 * half;
        const float ig = sigf(acc[0][v] + gv[0][v]);
        const float fg = sigf(acc[1][v] + gv[1][v]);
        const float gg = fast_tanhf(acc[2][v] + gv[2][v]);
        const float og = sigf(acc[3][v] + gv[3][v]);
        const float cn = fg * cst[v] + ig * gg;
        const float hn = og * fast_tanhf(cn);
        cst[v] = cn;
        h_s[m * HID + col] = hn;
        __builtin_nontemporal_store(hn, outrow + m * HID);
      }
    } else {
      #pragma unroll
      for (int v = 0; v < 8; ++v) {
        const int m = v + 8 * half;
        const float ig = sigf(acc[0][v] + gv[0][v]);
        const float fg = sigf(acc[1][v] + gv[1][v]);
        const float gg = fast_tanhf(acc[2][v] + gv[2][v]);
        const float og = sigf(acc[3][v] + gv[3][v]);
        const float cn = fg * cst[v] + ig * gg;
        const float hn = og * fast_tanhf(cn);
        if (m < nact) {
          cst[v] = cn;
          h_s[m * HID + col] = hn;
          __builtin_nontemporal_store(hn, outrow + m * HID);
        }
      }
    }
    off += bs_t;
    __syncthreads();  // h_s visible before next step's fragment reads
  }

  #pragma unroll
  for (int v = 0; v < 8; ++v) {
    const int m = v + 8 * half;
    c_fin[(r * 16 + m) * HID + col] = cst[v];
    h_fin[(r * 16 + m) * HID + col] = h_s[m * HID + col];
  }
}

// ---------------------------------------------------------------------------
// Path B fallback: fused kernel (both GEMMs per step). 16 WGs x 512 threads.
// ---------------------------------------------------------------------------
__global__ __launch_bounds__(512, 1) void lstm_fused(
    const float* __restrict__ data, const float* __restrict__ w_ih,
    const float* __restrict__ w_hh, const float* __restrict__ b_ih,
    const float* __restrict__ b_hh, const int* __restrict__ batch_sizes,
    float* __restrict__ packed_out, float* __restrict__ h_fin,
    float* __restrict__ c_fin)
{
  __shared__ float x_s[16 * INDIM];
  __shared__ float h_s[16 * HID];

  const int r    = blockIdx.x;
  const int tid  = threadIdx.x;
  const int j    = tid >> 5;         // wave id == H-column tile
  const int lane = tid & 31;
  const int ln16 = lane & 15;
  const int half = lane >> 4;
  const int col  = j * 16 + ln16;

  for (int i = tid; i < 16 * HID; i += 512) h_s[i] = 0.0f;

  float bias[4];
  #pragma unroll
  for (int g = 0; g < 4; ++g) {
    const int n = g * HID + col;
    bias[g] = b_ih[n] + b_hh[n];
  }
  v8f cst;
  #pragma unroll
  for (int v = 0; v < 8; ++v) cst[v] = 0.0f;

  const float* wi = w_ih + (size_t)col * INDIM + 2 * half;
  const float* wh = w_hh + (size_t)col * HID + 2 * half;

  long long off = 0;
  __syncthreads();

  for (int t = 0; t < TSTEPS; ++t) {
    const int bs_t = batch_sizes[t];
    const int act  = bs_t - r * 16;
    if (act <= 0) break;
    const int nact = act > 16 ? 16 : act;

    {  // x tile load: 512 threads x 8 floats
      const int m  = tid >> 5;
      const int c0 = (tid & 31) * 8;
      float4* dst = (float4*)&x_s[m * INDIM + c0];
      if (m < nact) {
        const float4* src =
            (const float4*)(data + (off + r * 16 + m) * (long long)INDIM + c0);
        dst[0] = src[0]; dst[1] = src[1];
      } else {
        float4 z = make_float4(0.f, 0.f, 0.f, 0.f);
        dst[0] = z; dst[1] = z;
      }
    }
    __syncthreads();

    v8f acc[4];
    #pragma unroll
    for (int g = 0; g < 4; ++g)
      #pragma unroll
      for (int v = 0; v < 8; ++v) acc[g][v] = bias[g];

    const float* xs = &x_s[ln16 * INDIM + 2 * half];
    const float* hs = &h_s[ln16 * HID + 2 * half];
    #pragma unroll 2
    for (int k = 0; k < INDIM; k += 4) {
      v2f a = *(const v2f*)(xs + k);
      #pragma unroll
      for (int g = 0; g < 4; ++g) {
        v2f b = *(const v2f*)(wi + (size_t)g * HID * INDIM + k);
        acc[g] = wmma4(a, b, acc[g]);
      }
    }
    #pragma unroll 2
    for (int k = 0; k < HID; k += 4) {
      v2f a = *(const v2f*)(hs + k);
      #pragma unroll
      for (int g = 0; g < 4; ++g) {
        v2f b = *(const v2f*)(wh + (size_t)g * HID * HID + k);
        acc[g] = wmma4(a, b, acc[g]);
      }
    }
    __syncthreads();

    float* outrow = packed_out + (off + r * 16) * (long long)HID + col;
    if (nact == 16) {
      #pragma unroll
      for (int v = 0; v < 8; ++v) {
        const int m = v + 8 * half;
        const float ig = sigf(acc[0][v]);
        const float fg = sigf(acc[1][v]);
        const float gg = fast_tanhf(acc[2][v]);
        const float og = sigf(acc[3][v]);
        const float cn = fg * cst[v] + ig * gg;
        const float hn = og * fast_tanhf(cn);
        cst[v] = cn;
        h_s[m * HID + col] = hn;
        __builtin_nontemporal_store(hn, outrow + m * HID);
      }
    } else {
      #pragma unroll
      for (int v = 0; v < 8; ++v) {
        const int m = v + 8 * half;
        const float ig = sigf(acc[0][v]);
        const float fg = sigf(acc[1][v]);
        const float gg = fast_tanhf(acc[2][v]);
        const float og = sigf(acc[3][v]);
        const float cn = fg * cst[v] + ig * gg;
        const float hn = og * fast_tanhf(cn);
        if (m < nact) {
          cst[v] = cn;
          h_s[m * HID + col] = hn;
          __builtin_nontemporal_store(hn, outrow + m * HID);
        }
      }
    }
    off += bs_t;
    __syncthreads();
  }

  #pragma unroll
  for (int v = 0; v < 8; ++v) {
    const int m = v + 8 * half;
    c_fin[(r * 16 + m) * HID + col] = cst[v];
    h_fin[(r * 16 + m) * HID + col] = h_s[m * HID + col];
  }
}

extern "C" void kernel_launch(void* const* d_in, const int* in_sizes, int n_in,
                              void* d_out, int out_size, void* d_ws, size_t ws_size,
                              hipStream_t stream) {
  // Inputs: data, w_ih, w_hh, b_ih, b_hh, batch_sizes, max_batch
  const float* data = (const float*)d_in[0];
  const float* w_ih = (const float*)d_in[1];
  const float* w_hh = (const float*)d_in[2];
  const float* b_ih = (const float*)d_in[3];
  const float* b_hh = (const float*)d_in[4];
  const int*   bs   = (const int*)d_in[5];

  const long long total = in_sizes[0] / INDIM;

  // d_out = [packed_out (total*256) | h_fin (256*256) | c_fin (256*256)]
  float* out   = (float*)d_out;
  float* h_fin = out + (size_t)total * HID;
  float* c_fin = h_fin + 256 * HID;

  const size_t need = (size_t)total * G_STRIDE * sizeof(float);
  if (ws_size >= need) {
    float* G = (float*)d_ws;
    dim3 grid((unsigned)((total + 15) / 16), 8);
    lstm_pregemm<<<grid, 256, 0, stream>>>(data, w_ih, b_ih, b_hh, G, total);
    lstm_rec<<<16, 512, 0, stream>>>(G, w_hh, bs, out, h_fin, c_fin, total);
  } else {
    lstm_fused<<<16, 512, 0, stream>>>(data, w_ih, w_hh, b_ih, b_hh, bs,
                                       out, h_fin, c_fin);
  }
}